// SpatialAttention_14267881357343
// MI455X (gfx1250) — compile-verified
//
#include <hip/hip_runtime.h>
#include <hip/hip_bf16.h>
#include <math.h>

#define NS 8192
#define CD 64   // C == N_H == 64

typedef float v2f __attribute__((ext_vector_type(2)));
typedef float v8f __attribute__((ext_vector_type(8)));

// ---------------------------------------------------------------------------
// Stage 1: X1[n,c] = (sum_c' H'[n,c',h] * w1[c']) @ W2^T ; X2[n,c] = sum_h H'[n,c,h] * w3[h]
// One 64-thread block per node n. H'[n] tile (64x64 f32 = 16KB) staged in LDS.
// ---------------------------------------------------------------------------
__global__ __launch_bounds__(64)
void stage1_kernel(const float* __restrict__ Hp,
                   const float* __restrict__ w1,
                   const float* __restrict__ W2,
                   const float* __restrict__ w3,
                   float* __restrict__ X1,
                   float* __restrict__ X2) {
    __shared__ float tile[64][65];   // [c][h], padded against bank conflicts
    __shared__ float sArr[64];
    const int n = blockIdx.x;
    const int t = threadIdx.x;
    const float* src = Hp + (size_t)n * 64 * 64;

    // coalesced cooperative load of H'[n] (4096 floats, float4 per thread x16)
    #pragma unroll
    for (int i = 0; i < 16; ++i) {
        int idx = i * 256 + t * 4;
        int r = idx >> 6, c = idx & 63;
        const float4 v = *reinterpret_cast<const float4*>(src + idx);
        tile[r][c + 0] = v.x; tile[r][c + 1] = v.y;
        tile[r][c + 2] = v.z; tile[r][c + 3] = v.w;
    }
    __syncthreads();

    // s[h] = sum_c tile[c][h] * w1[c]   (thread t == h)
    float s = 0.f;
    #pragma unroll 8
    for (int c = 0; c < 64; ++c) s += tile[c][t] * w1[c];
    sArr[t] = s;
    __syncthreads();

    // X1[n,c] = sum_h s[h] * W2[c,h]; X2[n,c] = sum_h tile[c][h] * w3[h]  (thread t == c)
    float x1 = 0.f, x2 = 0.f;
    const float* w2row = W2 + t * 64;
    #pragma unroll 8
    for (int h = 0; h < 64; ++h) {
        x1 += sArr[h] * w2row[h];
        x2 += tile[t][h] * w3[h];
    }
    X1[n * 64 + t] = x1;
    X2[n * 64 + t] = x2;
}

// ---------------------------------------------------------------------------
// Stage 2: fused S = X1 @ X2^T (WMMA f32 16x16x4), E = V*sigmoid(S+B),
// E_exp = exp(E)*mask written to out, row partial sums atomically accumulated.
// Block = 256 threads (8 waves): 128 rows x 64 cols of output per block.
// Tiles staged with GLOBAL_LOAD_ASYNC_TO_LDS_B128 (ASYNCcnt-tracked, no VGPR
// round-trip), then WMMA fragments read from LDS.
// ---------------------------------------------------------------------------
__global__ __launch_bounds__(256)
void gemm_edge_kernel(const float* __restrict__ X1,
                      const float* __restrict__ X2,
                      const float* __restrict__ V,
                      const float* __restrict__ Bm,
                      const unsigned char* __restrict__ mask,
                      float* __restrict__ out,
                      float* __restrict__ rowsum) {
    // row stride 68 floats: every row 16B-aligned (68*4=272=17*16), conflict-free frags
    __shared__ float X1s[128][68];
    __shared__ float X2s[64][68];

    const int tid  = threadIdx.x;
    const int wave = tid >> 5;
    const int lane = tid & 31;
    const int rowBase = blockIdx.y * 128;
    const int colBase = blockIdx.x * 64;

    // LDS byte offsets of the staging tiles (generic LDS addr low 32 bits = offset)
    const unsigned ldsX1 = (unsigned)(uintptr_t)(&X1s[0][0]);
    const unsigned ldsX2 = (unsigned)(uintptr_t)(&X2s[0][0]);

    // async-stage X1 tile: 128x64 f32 = 2048 x 16B chunks, 8 per thread
    #pragma unroll
    for (int i = 0; i < 8; ++i) {
        const int chunk = i * 256 + tid;
        const int r = chunk >> 4;            // row within tile
        const int c = (chunk & 15) * 4;      // float column
        const unsigned lds = ldsX1 + (unsigned)((r * 68 + c) * 4);
        const unsigned gof = (unsigned)(((rowBase + r) * 64 + c) * 4);
        asm volatile("global_load_async_to_lds_b128 %0, %1, %2"
                     :: "v"(lds), "v"(gof), "s"(X1) : "memory");
    }
    // async-stage X2 tile: 64x64 f32 = 1024 x 16B chunks, 4 per thread
    #pragma unroll
    for (int i = 0; i < 4; ++i) {
        const int chunk = i * 256 + tid;
        const int r = chunk >> 4;
        const int c = (chunk & 15) * 4;
        const unsigned lds = ldsX2 + (unsigned)((r * 68 + c) * 4);
        const unsigned gof = (unsigned)(((colBase + r) * 64 + c) * 4);
        asm volatile("global_load_async_to_lds_b128 %0, %1, %2"
                     :: "v"(lds), "v"(gof), "s"(X2) : "memory");
    }
    asm volatile("s_wait_asynccnt 0x0" ::: "memory");
    __syncthreads();

    const int m = lane & 15;   // row within A-frag / N within B-frag / N within D
    const int g = lane >> 4;   // half-wave selects K pair {0,1} vs {2,3}

    v8f acc[4] = {v8f{}, v8f{}, v8f{}, v8f{}};

    #pragma unroll
    for (int kc = 0; kc < 16; ++kc) {
        const int k = kc * 4 + 2 * g;
        v2f a;
        a.x = X1s[wave * 16 + m][k];
        a.y = X1s[wave * 16 + m][k + 1];
        #pragma unroll
        for (int nt = 0; nt < 4; ++nt) {
            v2f b;
            b.x = X2s[nt * 16 + m][k];
            b.y = X2s[nt * 16 + m][k + 1];
            acc[nt] = __builtin_amdgcn_wmma_f32_16x16x4_f32(
                false, a, false, b, (short)0, acc[nt], false, false);
        }
    }

    // fused epilogue: sigmoid gate, exp, mask, store, and row-sum reduction
    #pragma unroll
    for (int v = 0; v < 8; ++v) {
        const int r = rowBase + wave * 16 + v + 8 * g;   // global row of this D element
        float partial = 0.f;
        #pragma unroll
        for (int nt = 0; nt < 4; ++nt) {
            const int cg = colBase + nt * 16 + m;
            const size_t off = (size_t)r * NS + cg;
            const float sv = acc[nt][v] + Bm[off];
            const float e  = V[off] / (1.f + __expf(-sv));   // V * sigmoid(S+B)
            const float ee = __expf(e) * (mask[off] ? 1.f : 0.f);
            out[off] = ee;
            partial += ee;
        }
        // reduce over the 16 lanes (N dimension) inside each half-wave
        partial += __shfl_xor(partial, 1, 32);
        partial += __shfl_xor(partial, 2, 32);
        partial += __shfl_xor(partial, 4, 32);
        partial += __shfl_xor(partial, 8, 32);
        if (m == 0) atomicAdd(&rowsum[r], partial);
    }
}

// ---------------------------------------------------------------------------
// Stage 3: A = E_exp / (rowsum + 1e-6), in place on d_out.
// ---------------------------------------------------------------------------
__global__ __launch_bounds__(256)
void normalize_kernel(float* __restrict__ out, const float* __restrict__ rowsum) {
    const int row = blockIdx.y;
    const int col = (blockIdx.x * 256 + threadIdx.x) * 4;
    const float inv = 1.0f / (rowsum[row] + 1e-6f);
    float4* p = reinterpret_cast<float4*>(out + (size_t)row * NS + col);
    float4 vv = *p;
    vv.x *= inv; vv.y *= inv; vv.z *= inv; vv.w *= inv;
    *p = vv;
}

extern "C" void kernel_launch(void* const* d_in, const int* in_sizes, int n_in,
                              void* d_out, int out_size, void* d_ws, size_t ws_size,
                              hipStream_t stream) {
    (void)in_sizes; (void)n_in; (void)out_size; (void)ws_size;
    const float* Hp  = (const float*)d_in[0];           // [NS, 64, 64]
    const float* w1  = (const float*)d_in[1];           // [64]
    const float* W2  = (const float*)d_in[2];           // [64, 64]
    const float* w3  = (const float*)d_in[3];           // [64]
    const float* V   = (const float*)d_in[4];           // [NS, NS]
    const float* Bm  = (const float*)d_in[5];           // [NS, NS]
    const unsigned char* mask = (const unsigned char*)d_in[6]; // [NS, NS] bool
    float* out = (float*)d_out;

    // workspace layout: rowsum[NS] | X1[NS*64] | X2[NS*64]   (~4.2 MB)
    float* rowsum = (float*)d_ws;
    float* X1 = rowsum + NS;
    float* X2 = X1 + (size_t)NS * 64;

    hipMemsetAsync(rowsum, 0, NS * sizeof(float), stream);

    stage1_kernel<<<NS, 64, 0, stream>>>(Hp, w1, W2, w3, X1, X2);

    dim3 g2(NS / 64, NS / 128);   // (col blocks, row blocks)
    gemm_edge_kernel<<<g2, 256, 0, stream>>>(X1, X2, V, Bm, mask, out, rowsum);

    dim3 g3(NS / 1024, NS);
    normalize_kernel<<<g3, 256, 0, stream>>>(out, rowsum);
}